// GPDGaussian_27479200759872
// MI455X (gfx1250) — compile-verified
//
#include <hip/hip_runtime.h>

// Fused GPD-Gaussian kernel for MI455X (gfx1250, wave32).
// One wave = one pixel. Per-pixel 16x16x16 matmul S = (R*diag(d)) x R^T is done
// with 4 chained V_WMMA_F32_16X16X4_F32. Everything else is register-resident;
// rotation coefficients and diag entries are shared via v_readlane (no LDS).
//
// NOTE: all reads of the register array r[] use literal constant indices,
// hoisted into named scalars before any ternary select — this keeps r[] fully
// SROA-promoted and avoids v_cmp/v_cndmask dynamic-index ladders.

namespace {

constexpr int   kHW       = 64 * 64;
constexpr int   kB        = 4;
constexpr int   kNPix     = kB * kHW;            // 16384 pixels
constexpr int   kMElems   = kB * 16 * kHW;       // 262144  (m and s planes)
constexpr int   kMatElems = kNPix * 16 * 16;     // 4194304 (S, R, D planes)
constexpr float kPi       = 3.14159265358979323846f;

typedef __attribute__((ext_vector_type(2))) float v2f;
typedef __attribute__((ext_vector_type(8))) float v8f;

__device__ __forceinline__ float lane_bcast(float v, int srcLane) {
  return __builtin_bit_cast(
      float, __builtin_amdgcn_readlane(__builtin_bit_cast(int, v), srcLane));
}

__device__ __forceinline__ float fast_tanhf(float x) {
#if __has_builtin(__builtin_amdgcn_tanhf)
  return __builtin_amdgcn_tanhf(x);      // gfx1250 v_tanh_f32
#elif __has_builtin(__builtin_amdgcn_tanh_f32)
  return __builtin_amdgcn_tanh_f32(x);
#else
  return tanhf(x);
#endif
}

__global__ __launch_bounds__(256) void gpd_gaussian_kernel(
    const float* __restrict__ x, const float* __restrict__ Wenc,
    const float* __restrict__ benc, float* __restrict__ out) {
  const int lane = threadIdx.x & 31;
  const int wave = threadIdx.x >> 5;
  const int pix  = blockIdx.x * 8 + wave;        // grid sized so pix < kNPix
  const int b    = pix / kHW;
  const int hw   = pix - b * kHW;

  // x is (B,3,H,W); all lanes of the wave read the same pixel's 3 values.
  const float x0 = x[(b * 3 + 0) * kHW + hw];
  const float x1 = x[(b * 3 + 1) * kHW + hw];
  const float x2 = x[(b * 3 + 2) * kHW + hw];

  float* __restrict__ out_m = out;                    // (B,16,H,W)
  float* __restrict__ out_S = out + kMElems;          // (B,H,W,16,16)
  float* __restrict__ out_R = out_S + kMatElems;      // (B,H,W,16,16)
  float* __restrict__ out_D = out_R + kMatElems;      // (B,H,W,16,16)
  float* __restrict__ out_s = out_D + kMatElems;      // (B,16,H,W)

  const int  k  = lane & 15;       // row / channel owned by this lane
  const bool lo = lane < 16;       // lower half of the wave

  // ---- mean + scale channels (channel k; upper half replicates) ----------
  const float wm = Wenc[k * 3 + 0] * x0 + Wenc[k * 3 + 1] * x1 +
                   Wenc[k * 3 + 2] * x2 + benc[k];
  const int   os  = 16 + k;
  const float wsv = Wenc[os * 3 + 0] * x0 + Wenc[os * 3 + 1] * x1 +
                    Wenc[os * 3 + 2] * x2 + benc[os];
  const float sv  = 0.01f + 100.0f * (1.0f / (1.0f + __expf(-wsv)));
  const float dvl = 1.0f / (sv * sv);   // this lane's diag entry d[k]

  // ---- rotation angles: lane handles t = lane + 32*q ---------------------
  float ca[4], sa[4];
#pragma unroll
  for (int q = 0; q < 4; ++q) {
    const int   t   = lane + 32 * q;
    const int   tc  = t < 120 ? t : 119;          // clamp (dupes are unread)
    const int   o   = 32 + tc;
    const float wa  = Wenc[o * 3 + 0] * x0 + Wenc[o * 3 + 1] * x1 +
                      Wenc[o * 3 + 2] * x2 + benc[o];
    const float ang = kPi * fast_tanhf(wa);
    ca[q] = __cosf(ang);
    sa[q] = __sinf(ang);
  }

  // ---- R: identity row k, then 120 sequential Givens column rotations ----
  float r[16];
#pragma unroll
  for (int n = 0; n < 16; ++n) r[n] = (n == k) ? 1.0f : 0.0f;

  {
    int t = 0;
#pragma unroll
    for (int i = 0; i < 15; ++i) {
#pragma unroll
      for (int j = i + 1; j < 16; ++j) {
        const float c  = lane_bcast(ca[t >> 5], t & 31);  // t const after unroll
        const float s  = lane_bcast(sa[t >> 5], t & 31);
        const float ri = r[i], rj = r[j];
        r[i] = ri * c + rj * s;       // new col i
        r[j] = rj * c - ri * s;       // new col j
        ++t;
      }
    }
  }

  // ---- S = (R * diag(d)) x R^T via 4 chained V_WMMA_F32_16X16X4_F32 ------
  // A (16x4 f32) layout: VGPR0 = K0 (lanes 0-15) / K2 (lanes 16-31), VGPR1 = K1/K3.
  // B (4x16 f32) rows are R columns = each lane's own row registers, same split.
  // A is just B scaled columnwise by d.
  v8f acc = {0.f, 0.f, 0.f, 0.f, 0.f, 0.f, 0.f, 0.f};
#pragma unroll
  for (int tt = 0; tt < 4; ++tt) {
    const int   b0 = 4 * tt;
    const float d0 = lane_bcast(dvl, b0 + 0);
    const float d1 = lane_bcast(dvl, b0 + 1);
    const float d2 = lane_bcast(dvl, b0 + 2);
    const float d3 = lane_bcast(dvl, b0 + 3);
    // Hoist constant-index reads so r[] never sees a non-constant index.
    const float rk0 = r[b0 + 0];
    const float rk1 = r[b0 + 1];
    const float rk2 = r[b0 + 2];
    const float rk3 = r[b0 + 3];
    const float dx  = lo ? d0 : d2;
    const float dy  = lo ? d1 : d3;
    v2f A, Bv;
    Bv.x = lo ? rk0 : rk2;
    Bv.y = lo ? rk1 : rk3;
    A.x  = Bv.x * dx;
    A.y  = Bv.y * dy;
    acc = __builtin_amdgcn_wmma_f32_16x16x4_f32(false, A, false, Bv,
                                                (short)0, acc, false, false);
  }

  // ---- store S directly from the C/D layout ------------------------------
  // VGPR v, lanes 0-15: row v, col lane; lanes 16-31: row v+8, col lane-16.
  const long sbase  = (long)pix * 256;
  const int  rowoff = lo ? 0 : 8;
#pragma unroll
  for (int v = 0; v < 8; ++v) {
    out_S[sbase + (v + rowoff) * 16 + k] = acc[v];
  }

  // ---- lanes 0-15 store m, s, R (row-contiguous), D (diag) ---------------
  if (lo) {
    out_m[(b * 16 + k) * kHW + hw] = wm;
    out_s[(b * 16 + k) * kHW + hw] = sv;
#pragma unroll
    for (int n = 0; n < 16; n += 4) {
      float4 rv = make_float4(r[n], r[n + 1], r[n + 2], r[n + 3]);
      *reinterpret_cast<float4*>(out_R + sbase + k * 16 + n) = rv;
      float4 dm = make_float4(n + 0 == k ? dvl : 0.f, n + 1 == k ? dvl : 0.f,
                              n + 2 == k ? dvl : 0.f, n + 3 == k ? dvl : 0.f);
      *reinterpret_cast<float4*>(out_D + sbase + k * 16 + n) = dm;
    }
  }
}

}  // namespace

extern "C" void kernel_launch(void* const* d_in, const int* in_sizes, int n_in,
                              void* d_out, int out_size, void* d_ws,
                              size_t ws_size, hipStream_t stream) {
  const float* x    = (const float*)d_in[0];
  const float* Wenc = (const float*)d_in[1];
  const float* benc = (const float*)d_in[2];
  float*       out  = (float*)d_out;

  dim3 block(256);               // 8 wave32s -> 8 pixels per block
  dim3 grid(kNPix / 8);          // 2048 blocks
  gpd_gaussian_kernel<<<grid, block, 0, stream>>>(x, Wenc, benc, out);

  (void)in_sizes; (void)n_in; (void)out_size; (void)d_ws; (void)ws_size;
}